// MultiLayerLSTMParallel_43379169689744
// MI455X (gfx1250) — compile-verified
//
#include <hip/hip_runtime.h>
#include <hip/hip_bf16.h>

// ---------------------------------------------------------------------------
// Multi-layer parallel (log-space) LSTM for MI455X (gfx1250, wave32, WMMA).
// B=64, S=512, H=256, OUT=128, L=4.
// GEMMs via v_wmma_f32_16x16x4_f32; A-tile staging via Tensor Data Mover.
// ---------------------------------------------------------------------------

constexpr int kB   = 64;
constexpr int kS   = 512;
constexpr int kH   = 256;
constexpr int kOUT = 128;
constexpr int kM   = kB * kS;          // 32768 rows of activations
constexpr size_t kNel = (size_t)kM * kH;  // 8,388,608 elements per buffer

typedef __attribute__((ext_vector_type(2))) float v2f;
typedef __attribute__((ext_vector_type(8))) float v8f;
typedef __attribute__((ext_vector_type(4))) unsigned int v4u;
typedef __attribute__((ext_vector_type(8))) int v8i;
typedef __attribute__((ext_vector_type(4))) int v4i;

// D = A(16x4 f32) * B(4x16 f32) + C(16x16 f32), wave32 WMMA.
static __device__ __forceinline__ v8f wmma4(v2f a, v2f b, v8f c) {
    return __builtin_amdgcn_wmma_f32_16x16x4_f32(
        /*neg_a=*/false, a, /*neg_b=*/false, b,
        /*c_mod=*/(short)0, c, /*reuse_a=*/false, /*reuse_b=*/false);
}

// Numerically stable softplus: log(1 + e^t)
static __device__ __forceinline__ float sp(float t) {
    return fmaxf(t, 0.0f) + log1pf(expf(-fabsf(t)));
}
// log_g(z) = z>=0 ? log(relu(z)+0.5) : -softplus(-z)
static __device__ __forceinline__ float log_g(float z) {
    return (z >= 0.0f) ? logf(z + 0.5f) : -sp(-z);
}
static __device__ __forceinline__ float logaddexp(float x, float y) {
    float m = fmaxf(x, y);
    return m + log1pf(expf(-fabsf(x - y)));
}
// Fused gate epilogue: pre-activations -> (log_f, v = log_i + log_g(ph))
static __device__ __forceinline__ void gate_math(float pf, float pi, float ph,
                                                 float& lf, float& vv) {
    float diff = sp(-pf) - sp(-pi);
    lf = -sp(diff);
    vv = -sp(-diff) + log_g(ph);
}

// ---------------------------------------------------------------------------
// Layer 0: din == 1 -> gates are rank-1; elementwise over (b*s, h).
// ---------------------------------------------------------------------------
__global__ __launch_bounds__(256) void gates_l0(
    const float* __restrict__ x,
    const float* __restrict__ Wi, const float* __restrict__ bi,
    const float* __restrict__ Wf, const float* __restrict__ bf,
    const float* __restrict__ Wh, const float* __restrict__ bh,
    float* __restrict__ logf_o, float* __restrict__ v_o) {
    size_t i = (size_t)blockIdx.x * blockDim.x + threadIdx.x; // [0, M*H)
    int h  = (int)(i & (kH - 1));
    size_t bs = i >> 8;                                        // H == 256
    float xv = x[bs];
    float pf = fmaf(xv, Wf[h], bf[h]);
    float pi = fmaf(xv, Wi[h], bi[h]);
    float ph = fmaf(xv, Wh[h], bh[h]);
    float lf, vv;
    gate_math(pf, pi, ph, lf, vv);
    logf_o[i] = lf;
    v_o[i]    = vv;
}

// ---------------------------------------------------------------------------
// Layers 1..3 gate GEMM: P = A(32768x256) @ W^T(256x256) + b, 3 gates fused.
// Block = 128 threads (4 waves). Block tile: 64(M) x 64(N), all 3 gates.
// A staged into LDS by the Tensor Data Mover (2D tile DMA with native row
// padding: 128 DWORDs data + 4 DWORDs pad -> LDA=132, conflict-free ds reads).
// B read from global (weights are L2-resident).
// ---------------------------------------------------------------------------
__global__ __launch_bounds__(128) void gates_gemm(
    const float* __restrict__ Ain,
    const float* __restrict__ Wi, const float* __restrict__ bi,
    const float* __restrict__ Wf, const float* __restrict__ bf,
    const float* __restrict__ Wh, const float* __restrict__ bh,
    float* __restrict__ logf_o, float* __restrict__ v_o) {
    constexpr int LDA = 132; // 128 + 4 pad DWORDs, produced by TDM pad engine
    __shared__ float As[64 * LDA]; // only __shared__ in kernel -> LDS offset 0

    const int m0 = blockIdx.x * 64;
    const int n0 = blockIdx.y * 64;
    const int tid  = threadIdx.x;
    const int wv   = tid >> 5;
    const int lane = tid & 31;
    const int half = lane >> 4;
    const int lrow = lane & 15;

    v8f acc[4][3] = {};

    // Warm the weight tiles for this block's N range into cache.
    if (tid < 4) {
        const int wr = (n0 + tid * 16) * kH;
        __builtin_prefetch(Wf + wr, 0, 0);
        __builtin_prefetch(Wi + wr, 0, 0);
        __builtin_prefetch(Wh + wr, 0, 0);
    }

    for (int kc = 0; kc < kH; kc += 128) {
        if (wv == 0) {
            // Tensor DMA descriptor (D#): 64 rows x 128 DWORDs, row stride
            // 256 DWORDs in global, +4 pad DWORDs per row into LDS.
            unsigned long long gaddr =
                (unsigned long long)(const void*)(Ain + (size_t)m0 * kH + kc);
            v4u g0;
            g0.x = 1u;                                   // count=1, user mode
            g0.y = 0u;                                   // lds_addr = 0 (As)
            g0.z = (unsigned)(gaddr & 0xFFFFFFFFu);      // global_addr[31:0]
            g0.w = (unsigned)((gaddr >> 32) & 0x1FFFFFFu)// global_addr[56:32]
                 | (2u << 30);                           // type = 2 ("image")
            v8i g1;
            g1[0] = (int)((2u << 16)    // data_size = 4B
                        | (1u << 20)    // pad_enable
                        | (6u << 22)    // pad_interval: 128 DWORDs
                        | (3u << 25));  // pad_amount:   4 DWORDs
            g1[1] = (int)(256u << 16);  // tensor_dim0 = 256 (lo16); abar=0
            g1[2] = (int)(32768u << 16);// tensor_dim0 hi=0; tensor_dim1 lo=32768
            g1[3] = (int)(128u << 16);  // tensor_dim1 hi=0; tile_dim0 = 128
            g1[4] = 64;                 // tile_dim1 = 64; tile_dim2 = 0
            g1[5] = 256;                // tensor_dim0_stride = 256 (lo32)
            g1[6] = 0;                  // stride hi; tensor_dim1_stride lo
            g1[7] = 0;
            __builtin_amdgcn_tensor_load_to_lds(g0, g1, (v4i)0, (v4i)0,
                                                (v8i)0, 0);
            __builtin_amdgcn_s_wait_tensorcnt(0);
        }
        __syncthreads();

        const int arow = wv * 16 + lrow;
        for (int k = 0; k < 128; k += 4) {
            v2f a = *(const v2f*)(&As[arow * LDA + k + 2 * half]);
#pragma unroll
            for (int nt = 0; nt < 4; ++nt) {
                size_t off = (size_t)(n0 + nt * 16 + lrow) * kH + kc + k + 2 * half;
                v2f bfr = *(const v2f*)(Wf + off);
                v2f bir = *(const v2f*)(Wi + off);
                v2f bhr = *(const v2f*)(Wh + off);
                acc[nt][0] = wmma4(a, bfr, acc[nt][0]);
                acc[nt][1] = wmma4(a, bir, acc[nt][1]);
                acc[nt][2] = wmma4(a, bhr, acc[nt][2]);
            }
        }
        __syncthreads();
    }

    // Fused epilogue: bias add + log-space gate math + store log_f / v.
#pragma unroll
    for (int nt = 0; nt < 4; ++nt) {
        int col = n0 + nt * 16 + lrow;
        float bfv = bf[col], biv = bi[col], bhv = bh[col];
#pragma unroll
        for (int r = 0; r < 8; ++r) {
            int row = m0 + wv * 16 + r + 8 * half;
            float lf, vv;
            gate_math(acc[nt][0][r] + bfv, acc[nt][1][r] + biv,
                      acc[nt][2][r] + bhv, lf, vv);
            size_t oi = (size_t)row * kH + col;
            logf_o[oi] = lf;
            v_o[oi]    = vv;
        }
    }
}

// ---------------------------------------------------------------------------
// Log-space scan over time + sigmoid. One thread per (b, h) chain.
//   L_0 = log_g(0) = log(0.5);  L_s = logaddexp(log_f_s + L_{s-1}, v_s)
//   h_s = sigmoid(exp(L_s))
// Coalesced over h (consecutive threads -> consecutive h).
// ---------------------------------------------------------------------------
__global__ __launch_bounds__(256) void scan_kernel(
    const float* __restrict__ logf_i, const float* __restrict__ v_i,
    float* __restrict__ hout) {
    int i = blockIdx.x * blockDim.x + threadIdx.x; // [0, B*H)
    int b = i >> 8;
    int h = i & (kH - 1);
    float L = -0.69314718056f; // log(0.5)
    size_t base = (size_t)b * kS * kH + h;
    for (int s = 0; s < kS; ++s) {
        size_t idx = base + (size_t)s * kH;
        float lf = logf_i[idx];
        float vv = v_i[idx];
        L = logaddexp(lf + L, vv);
        float e = expf(L);
        hout[idx] = 1.0f / (1.0f + expf(-e));
    }
}

// ---------------------------------------------------------------------------
// Output projection: (64 x 131072) @ (131072 x 128) + bout.
// 32 blocks (4 M-tiles x 8 N-tiles), 8 waves/block, K split across waves,
// WMMA accumulate, LDS reduction, bias add. No atomics (deterministic).
// ---------------------------------------------------------------------------
__global__ __launch_bounds__(256) void out_gemm(
    const float* __restrict__ hbuf, const float* __restrict__ Wout,
    const float* __restrict__ bout, float* __restrict__ out) {
    __shared__ float red[8 * 256];
    const int tid  = threadIdx.x;
    const int wv   = tid >> 5;
    const int lane = tid & 31;
    const int half = lane >> 4;
    const int lrow = lane & 15;
    const int mt    = blockIdx.x >> 3;
    const int ntile = blockIdx.x & 7;
    const int m0 = mt * 16;
    const int n0 = ntile * 16;
    const size_t K = (size_t)kS * kH;      // 131072
    const size_t Kslice = K / 8;           // 16384 per wave

    v8f acc = {};
    const float* aP = hbuf + (size_t)(m0 + lrow) * K + (size_t)wv * Kslice + 2 * half;
    const float* bP = Wout + (size_t)(n0 + lrow) * K + (size_t)wv * Kslice + 2 * half;
    __builtin_prefetch(bP, 0, 0);
    for (size_t k = 0; k < Kslice; k += 4) {
        v2f a = *(const v2f*)(aP + k);
        v2f b = *(const v2f*)(bP + k);
        acc = wmma4(a, b, acc);
    }
#pragma unroll
    for (int r = 0; r < 8; ++r) red[wv * 256 + r * 32 + lane] = acc[r];
    __syncthreads();

    // Each thread owns one accumulator slot; sum the 8 wave partials.
    int slot = tid;
    float t = 0.0f;
#pragma unroll
    for (int w = 0; w < 8; ++w) t += red[w * 256 + slot];
    int r  = slot >> 5;
    int ln = slot & 31;
    int row = m0 + r + 8 * (ln >> 4);
    int col = n0 + (ln & 15);
    out[row * kOUT + col] = t + bout[col];
}

// ---------------------------------------------------------------------------
// Host launcher. Workspace: h (33.5MB) | log_f (33.5MB) | v (33.5MB).
// The scan only reads log_f/v, so h may be overwritten in place each layer.
// ---------------------------------------------------------------------------
extern "C" void kernel_launch(void* const* d_in, const int* in_sizes, int n_in,
                              void* d_out, int out_size, void* d_ws, size_t ws_size,
                              hipStream_t stream) {
    (void)in_sizes; (void)n_in; (void)out_size; (void)ws_size;
    const float* x    = (const float*)d_in[0];
    const float* Wout = (const float*)d_in[25];
    const float* bout = (const float*)d_in[26];

    float* ws   = (float*)d_ws;
    float* hbuf = ws;
    float* lgf  = ws + kNel;
    float* vb   = ws + 2 * kNel;

    // Per-layer params: dict insertion order is Wi, bi, Wf, bf, Wh, bh.
    auto P = [&](int l, int j) { return (const float*)d_in[1 + 6 * l + j]; };

    // Layer 0 (din == 1): elementwise gates, then scan.
    gates_l0<<<(unsigned)(kNel / 256), 256, 0, stream>>>(
        x, P(0,0), P(0,1), P(0,2), P(0,3), P(0,4), P(0,5), lgf, vb);
    scan_kernel<<<kB * kH / 256, 256, 0, stream>>>(lgf, vb, hbuf);

    // Layers 1..3: WMMA gate GEMMs + scan (h overwritten in place).
    dim3 ggrid(kM / 64, kH / 64);
    for (int l = 1; l < 4; ++l) {
        gates_gemm<<<ggrid, 128, 0, stream>>>(
            hbuf, P(l,0), P(l,1), P(l,2), P(l,3), P(l,4), P(l,5), lgf, vb);
        scan_kernel<<<kB * kH / 256, 256, 0, stream>>>(lgf, vb, hbuf);
    }

    // Final projection.
    out_gemm<<<(kB / 16) * (kOUT / 16), 256, 0, stream>>>(hbuf, Wout, bout,
                                                          (float*)d_out);
}